// RNNWithMemoryEncoder_55740085567673
// MI455X (gfx1250) — compile-verified
//
#include <hip/hip_runtime.h>

// ---------------------------------------------------------------------------
// RNNWithMemoryEncoder for MI455X (gfx1250, wave32, WMMA + TDM).
// Shapes fixed by reference: B=32, T=128, L=512, M=4, D=256, HOPS=3, fp32.
// All GEMMs use V_WMMA_F32_16X16X4_F32 (fp32-exact matrix path).
// GRU gate inputs (gx rows, 96KB/step) are staged into LDS by the Tensor
// Data Mover, overlapped with the recurrent WMMA GEMM, synced by TENSORcnt.
// ---------------------------------------------------------------------------

#define Bsz   32
#define Tlen  128
#define Llen  512
#define Mbag  4
#define Dh    256
#define D3    768
#define VOC   32000

typedef __attribute__((ext_vector_type(2))) float v2f;
typedef __attribute__((ext_vector_type(8))) float v8f;

__device__ __forceinline__ v8f wmma4(v2f a, v2f b, v8f c) {
  // 8 args: (neg_a, A, neg_b, B, c_mod, C, reuse_a, reuse_b)
  return __builtin_amdgcn_wmma_f32_16x16x4_f32(false, a, false, b, (short)0, c,
                                               false, false);
}

__device__ __forceinline__ float sigmoidf_(float x) {
  return 1.0f / (1.0f + __expf(-x));
}

// ---------------------------------------------------------------------------
// Tensor Data Mover: 1-D tile (nelem fp32, nelem <= 65535) global -> LDS.
// D# layout per CDNA5 ISA (groups 0/1; 2/3 zero).
// This toolchain (clang-23) declares the 6-arg builtin:
//   (uint32x4 g0, int32x8 g1, int32x4 g2, int32x4 g3, int32x8, i32 cpol)
// ---------------------------------------------------------------------------
#if __has_builtin(__builtin_amdgcn_tensor_load_to_lds)
#define USE_TDM 1
typedef __attribute__((ext_vector_type(4))) unsigned tdm_v4u;
typedef __attribute__((ext_vector_type(8))) int tdm_v8i;
typedef __attribute__((ext_vector_type(4))) int tdm_v4i;

__device__ __forceinline__ unsigned lds_addr_u32(const void* p) {
  // LDS aperture: flat addr[31:0] == workgroup-relative LDS byte offset.
  return (unsigned)(unsigned long long)(uintptr_t)p;
}

__device__ __forceinline__ void tdm_load_row_f32(const float* gsrc,
                                                 unsigned lds_off,
                                                 unsigned nelem) {
  const unsigned long long ga = (unsigned long long)(uintptr_t)gsrc;
  tdm_v4u g0;
  g0[0] = 1u;                                       // count=1, user mode
  g0[1] = lds_off;                                  // lds_addr (bytes)
  g0[2] = (unsigned)ga;                             // global_addr[31:0]
  g0[3] = (unsigned)((ga >> 32) & 0x1FFFFFFu)       // global_addr[56:32]
          | (2u << 30);                             // type=2 ("image")
  tdm_v8i g1;
  g1[0] = 0x00020000;                               // wg_mask=0, data_size=4B
  g1[1] = (int)((nelem & 0xFFFFu) << 16);           // tensor_dim0[15:0]
  g1[2] = (int)((nelem >> 16) & 0xFFFFu);           // tensor_dim0[31:16]
  g1[3] = (int)((nelem & 0xFFFFu) << 16);           // tile_dim0
  g1[4] = 0;                                        // tile_dim1/2 unused
  g1[5] = (int)nelem;                               // tensor_dim0_stride lo
  g1[6] = 0;
  g1[7] = 0;
  const tdm_v4i gz4 = {0, 0, 0, 0};
  const tdm_v8i gz8 = {0, 0, 0, 0, 0, 0, 0, 0};
  __builtin_amdgcn_tensor_load_to_lds(g0, g1, gz4, gz4, gz8, 0);
}
#else
#define USE_TDM 0
#endif

// ---------------------------------------------------------------------------
// 1) EmbeddingBag(sum) over conv_seqs -> emb stored [T,B,D] (GRU time-major).
// ---------------------------------------------------------------------------
__global__ void embed_conv_kernel(const int* __restrict__ conv_seqs,
                                  const float* __restrict__ emb_ctx,
                                  float* __restrict__ emb_out) {
  const int bt = blockIdx.x;            // 0..B*T-1, bt = b*T + t
  const int b = bt >> 7;
  const int t = bt & (Tlen - 1);
  const int d = threadIdx.x;            // 0..255
  const int* idx = conv_seqs + (size_t)bt * Mbag;
  float s = emb_ctx[(size_t)idx[0] * Dh + d] + emb_ctx[(size_t)idx[1] * Dh + d] +
            emb_ctx[(size_t)idx[2] * Dh + d] + emb_ctx[(size_t)idx[3] * Dh + d];
  emb_out[((size_t)t * Bsz + b) * Dh + d] = s;
}

// ---------------------------------------------------------------------------
// 2) gx = emb[T*B,256] @ Wih[768,256]^T + bih, for both directions.
//    WMMA f32 16x16x4; one 16x16 output tile per wave; K-loop of 64 steps.
//    A/B lane layout: row = lane&15, k-pair = (lane>>4)*2 (ISA 7.12.2).
// ---------------------------------------------------------------------------
__global__ void __launch_bounds__(256) gx_gemm_kernel(
    const float* __restrict__ emb, const float* __restrict__ Wih_f,
    const float* __restrict__ bih_f, const float* __restrict__ Wih_b,
    const float* __restrict__ bih_b, float* __restrict__ gx_f,
    float* __restrict__ gx_b) {
  const int wave = threadIdx.x >> 5;
  const int lane = threadIdx.x & 31;
  const int tile = blockIdx.x * 8 + wave;   // 256 mtiles * 48 ntiles = 12288
  const int NT = D3 / 16;                   // 48
  const int mtile = tile / NT;
  const int ntile = tile % NT;
  const float* W = blockIdx.y ? Wih_b : Wih_f;
  const float* bi = blockIdx.y ? bih_b : bih_f;
  float* gx = blockIdx.y ? gx_b : gx_f;

  const int r = lane & 15;
  const int koff = (lane >> 4) * 2;
  const float* Ap = emb + ((size_t)(mtile * 16 + r)) * Dh + koff;
  const float* Bp = W + ((size_t)(ntile * 16 + r)) * Dh + koff;

  v8f acc = {};
  for (int k = 0; k < Dh; k += 4) {
    v2f a = *(const v2f*)(Ap + k);
    v2f w = *(const v2f*)(Bp + k);
    acc = wmma4(a, w, acc);
  }
  const int col = ntile * 16 + (lane & 15);
  const float bias = bi[col];
  const int rowb = mtile * 16 + (lane >> 4) * 8;
#pragma unroll
  for (int v = 0; v < 8; ++v)
    gx[(size_t)(rowb + v) * D3 + col] = acc[v] + bias;
}

// ---------------------------------------------------------------------------
// 3) GRU recurrence. One block per direction.
//    LDS: h[32,256] (32KB) + gh[32,768] (96KB) + gx stage [32,768] (96KB).
//    Per step: gh = h @ Whh^T via WMMA (16 waves x (2 mtiles x 3 ntiles));
//    meanwhile the TDM streams the step's gx row into LDS (TENSORcnt sync);
//    then fused gate elementwise; backward dir stores ys time-reversed.
// ---------------------------------------------------------------------------
__global__ void __launch_bounds__(512) gru_kernel(
    const float* __restrict__ gx_f, const float* __restrict__ gx_b,
    const float* __restrict__ Whh_f, const float* __restrict__ bhh_f,
    const float* __restrict__ Whh_b, const float* __restrict__ bhh_b,
    float* __restrict__ ys_f, float* __restrict__ ys_b) {
  extern __shared__ float smem[];
  float* h = smem;                        // [32][256]
  float* gh = smem + Bsz * Dh;            // [32][768]
  float* gxlds = gh + Bsz * D3;           // [32][768] staged gx row

  const int dir = blockIdx.x;
  const float* gx = dir ? gx_b : gx_f;
  const float* Whh = dir ? Whh_b : Whh_f;
  const float* bhh = dir ? bhh_b : bhh_f;
  float* ys = dir ? ys_b : ys_f;

#if USE_TDM
  const unsigned gxlds_off = lds_addr_u32(gxlds);
  if (threadIdx.x < 32) {   // wave 0 issues; EXEC ignored by TDM, branch is wave-uniform
    const int t0 = dir ? (Tlen - 1) : 0;
    tdm_load_row_f32(gx + (size_t)t0 * Bsz * D3, gxlds_off, Bsz * D3);
  }
#endif

  for (int i = threadIdx.x; i < Bsz * Dh; i += blockDim.x) h[i] = 0.0f;
  __syncthreads();

  const int wave = threadIdx.x >> 5;      // 0..15
  const int lane = threadIdx.x & 31;
  const int r = lane & 15;
  const int koff = (lane >> 4) * 2;
  const int nt0 = 3 * wave;               // 3 n-tiles per wave -> 48 total

  const float* B0 = Whh + ((size_t)((nt0 + 0) * 16 + r)) * Dh + koff;
  const float* B1 = Whh + ((size_t)((nt0 + 1) * 16 + r)) * Dh + koff;
  const float* B2 = Whh + ((size_t)((nt0 + 2) * 16 + r)) * Dh + koff;
  const float* A0 = h + (size_t)r * Dh + koff;          // mtile 0 (rows 0..15)
  const float* A1 = h + (size_t)(16 + r) * Dh + koff;   // mtile 1 (rows 16..31)

  for (int s = 0; s < Tlen; ++s) {
    v8f acc00 = {}, acc01 = {}, acc02 = {};
    v8f acc10 = {}, acc11 = {}, acc12 = {};
    for (int k = 0; k < Dh; k += 4) {
      v2f a0 = *(const v2f*)(A0 + k);
      v2f a1 = *(const v2f*)(A1 + k);
      v2f b0 = *(const v2f*)(B0 + k);
      v2f b1 = *(const v2f*)(B1 + k);
      v2f b2 = *(const v2f*)(B2 + k);
      acc00 = wmma4(a0, b0, acc00);
      acc10 = wmma4(a1, b0, acc10);
      acc01 = wmma4(a0, b1, acc01);
      acc11 = wmma4(a1, b1, acc11);
      acc02 = wmma4(a0, b2, acc02);
      acc12 = wmma4(a1, b2, acc12);
    }
    // spill gate pre-activations to LDS (D layout: row = v + (lane>>4)*8)
    {
      const int rb0 = (lane >> 4) * 8;
      const int rb1 = 16 + rb0;
      const int c0 = (nt0 + 0) * 16 + r;
      const int c1 = (nt0 + 1) * 16 + r;
      const int c2 = (nt0 + 2) * 16 + r;
#pragma unroll
      for (int v = 0; v < 8; ++v) {
        gh[(rb0 + v) * D3 + c0] = acc00[v];
        gh[(rb1 + v) * D3 + c0] = acc10[v];
        gh[(rb0 + v) * D3 + c1] = acc01[v];
        gh[(rb1 + v) * D3 + c1] = acc11[v];
        gh[(rb0 + v) * D3 + c2] = acc02[v];
        gh[(rb1 + v) * D3 + c2] = acc12[v];
      }
    }
#if USE_TDM
    if (threadIdx.x < 32) __builtin_amdgcn_s_wait_tensorcnt(0);
#endif
    __syncthreads();   // gh visible; staged gx row published to all waves

    const int t_eff = dir ? (Tlen - 1 - s) : s;
#if USE_TDM
    const float* gxrow = gxlds;
#else
    const float* gxrow = gx + (size_t)t_eff * Bsz * D3;
#endif
    for (int i = threadIdx.x; i < Bsz * Dh; i += blockDim.x) {
      const int b = i >> 8;
      const int d = i & (Dh - 1);
      const float xr = gxrow[b * D3 + d];
      const float xz = gxrow[b * D3 + Dh + d];
      const float xn = gxrow[b * D3 + 2 * Dh + d];
      const float hr = gh[b * D3 + d] + bhh[d];
      const float hz = gh[b * D3 + Dh + d] + bhh[Dh + d];
      const float hn = gh[b * D3 + 2 * Dh + d] + bhh[2 * Dh + d];
      const float rg = sigmoidf_(xr + hr);
      const float zg = sigmoidf_(xz + hz);
      const float ng = tanhf(xn + rg * hn);
      const float h2 = (1.0f - zg) * ng + zg * h[i];
      ys[((size_t)t_eff * Bsz + b) * Dh + d] = h2;
      h[i] = h2;   // only this thread touches h[i] between barriers
    }
    __syncthreads();   // h updated; gx stage buffer free for next step
#if USE_TDM
    if (threadIdx.x < 32 && s + 1 < Tlen) {
      const int tn = dir ? (Tlen - 2 - s) : (s + 1);
      tdm_load_row_f32(gx + (size_t)tn * Bsz * D3, gxlds_off, Bsz * D3);
    }
#endif
  }
}

// ---------------------------------------------------------------------------
// 4) rnn_out[b,t,:] = cat(ys_f[t,b], ys_b[t,b]) @ W_w[256,512]^T + W_b.
//    M=4096 (r=b*128+t), N=256, K=512 (split across the two halves).
// ---------------------------------------------------------------------------
__global__ void __launch_bounds__(256) rnnout_gemm_kernel(
    const float* __restrict__ ys_f, const float* __restrict__ ys_b,
    const float* __restrict__ W_w, const float* __restrict__ W_b,
    float* __restrict__ rnn_out) {
  const int wave = threadIdx.x >> 5;
  const int lane = threadIdx.x & 31;
  const int tile = blockIdx.x * 8 + wave;  // 256 mtiles * 16 ntiles = 4096
  const int mtile = tile >> 4;
  const int ntile = tile & 15;
  const int r = lane & 15;
  const int koff = (lane >> 4) * 2;

  const int row = mtile * 16 + r;          // = b*T + t
  const int b = row >> 7;
  const int t = row & (Tlen - 1);
  const float* Af = ys_f + ((size_t)t * Bsz + b) * Dh + koff;
  const float* Ab = ys_b + ((size_t)t * Bsz + b) * Dh + koff;
  const float* Bp = W_w + ((size_t)(ntile * 16 + r)) * (2 * Dh) + koff;

  v8f acc = {};
  for (int k = 0; k < Dh; k += 4)
    acc = wmma4(*(const v2f*)(Af + k), *(const v2f*)(Bp + k), acc);
  for (int k = 0; k < Dh; k += 4)
    acc = wmma4(*(const v2f*)(Ab + k), *(const v2f*)(Bp + Dh + k), acc);

  const int col = ntile * 16 + (lane & 15);
  const float bias = W_b[col];
  const int rowb = mtile * 16 + (lane >> 4) * 8;
#pragma unroll
  for (int v = 0; v < 8; ++v)
    rnn_out[(size_t)(rowb + v) * Dh + col] = acc[v] + bias;  // [B,T,D] flat
}

// ---------------------------------------------------------------------------
// 5/7) out[32,256] = act( cat(A0,A1)[32,512] @ W[256,512]^T + bias ).
//     Used for `hidden` (no relu) and `encoded` (relu).
// ---------------------------------------------------------------------------
__global__ void __launch_bounds__(256) cat_gemm32_kernel(
    const float* __restrict__ A0, const float* __restrict__ A1,
    const float* __restrict__ W, const float* __restrict__ bias,
    float* __restrict__ out, int relu) {
  const int wave = threadIdx.x >> 5;
  const int lane = threadIdx.x & 31;
  const int tile = blockIdx.x * 8 + wave;  // 2 mtiles * 16 ntiles = 32
  const int mtile = tile >> 4;
  const int ntile = tile & 15;
  const int r = lane & 15;
  const int koff = (lane >> 4) * 2;

  const float* a0 = A0 + ((size_t)(mtile * 16 + r)) * Dh + koff;
  const float* a1 = A1 + ((size_t)(mtile * 16 + r)) * Dh + koff;
  const float* bp = W + ((size_t)(ntile * 16 + r)) * (2 * Dh) + koff;

  v8f acc = {};
  for (int k = 0; k < Dh; k += 4)
    acc = wmma4(*(const v2f*)(a0 + k), *(const v2f*)(bp + k), acc);
  for (int k = 0; k < Dh; k += 4)
    acc = wmma4(*(const v2f*)(a1 + k), *(const v2f*)(bp + Dh + k), acc);

  const int col = ntile * 16 + (lane & 15);
  const float bb = bias[col];
  const int rowb = mtile * 16 + (lane >> 4) * 8;
#pragma unroll
  for (int v = 0; v < 8; ++v) {
    float x = acc[v] + bb;
    if (relu) x = fmaxf(x, 0.0f);
    out[(size_t)(rowb + v) * Dh + col] = x;
  }
}

// ---------------------------------------------------------------------------
// 6) Fused 3-hop memory network. One block per batch element; u, logits,
//    gathered indices cached in LDS. Writes sigmoid(final logits) to d_out
//    and the final query u to workspace.
// ---------------------------------------------------------------------------
__device__ __forceinline__ float dot4_(float4 a, float4 b) {
  return a.x * b.x + a.y * b.y + a.z * b.z + a.w * b.w;
}

__global__ void __launch_bounds__(256) hops_kernel(
    const int* __restrict__ src_seqs, const int* __restrict__ kb_len,
    const int* __restrict__ conv_len, const float* __restrict__ C_tables,
    const float* __restrict__ rnn_out, const float* __restrict__ hidden,
    float* __restrict__ u_out, float* __restrict__ gp_out) {
  const int b = blockIdx.x;
  const int tid = threadIdx.x;

  __shared__ float u[Dh];
  __shared__ float lg[Llen];
  __shared__ int sidx[Llen * Mbag];
  __shared__ int srel[Llen];
  __shared__ float red[256];

  u[tid] = hidden[(size_t)b * Dh + tid];
  const int kb = kb_len[b];
  const int cl = conv_len[b];
  for (int l = tid; l < Llen; l += 256) {
    const int rel = l - kb;
    srel[l] = (rel >= 0 && rel < cl) ? rel : -1;
    const int* ip = src_seqs + ((size_t)b * Llen + l) * Mbag;
    sidx[l * 4 + 0] = ip[0];
    sidx[l * 4 + 1] = ip[1];
    sidx[l * 4 + 2] = ip[2];
    sidx[l * 4 + 3] = ip[3];
  }
  __syncthreads();

  for (int hop = 0; hop < 3; ++hop) {
    const float* CA = C_tables + (size_t)hop * VOC * Dh;
    const float* CC = C_tables + (size_t)(hop + 1) * VOC * Dh;

    // logits[l] = <embed_A(b,l,:), u>
    for (int l = tid; l < Llen; l += 256) {
      const float4* uv = (const float4*)u;
      const float4* p0 = (const float4*)(CA + (size_t)sidx[l * 4 + 0] * Dh);
      const float4* p1 = (const float4*)(CA + (size_t)sidx[l * 4 + 1] * Dh);
      const float4* p2 = (const float4*)(CA + (size_t)sidx[l * 4 + 2] * Dh);
      const float4* p3 = (const float4*)(CA + (size_t)sidx[l * 4 + 3] * Dh);
      const int rel = srel[l];
      const float4* pr =
          (const float4*)(rnn_out + ((size_t)b * Tlen + (rel < 0 ? 0 : rel)) * Dh);
      float acc = 0.0f;
      for (int q = 0; q < Dh / 4; ++q) {
        const float4 uu = uv[q];
        acc += dot4_(uu, p0[q]) + dot4_(uu, p1[q]) + dot4_(uu, p2[q]) +
               dot4_(uu, p3[q]);
        if (rel >= 0) acc += dot4_(uu, pr[q]);
      }
      lg[l] = acc;
    }
    __syncthreads();

    if (hop == 2) {
      for (int l = tid; l < Llen; l += 256)
        gp_out[(size_t)b * Llen + l] = sigmoidf_(lg[l]);
    }

    // softmax over 512 entries
    red[tid] = fmaxf(lg[tid], lg[tid + 256]);
    for (int st = 128; st > 0; st >>= 1) {
      __syncthreads();
      if (tid < st) red[tid] = fmaxf(red[tid], red[tid + st]);
    }
    __syncthreads();
    const float mx = red[0];
    __syncthreads();
    const float e0 = __expf(lg[tid] - mx);
    const float e1 = __expf(lg[tid + 256] - mx);
    red[tid] = e0 + e1;
    for (int st = 128; st > 0; st >>= 1) {
      __syncthreads();
      if (tid < st) red[tid] += red[tid + st];
    }
    __syncthreads();
    const float inv = 1.0f / red[0];
    __syncthreads();
    lg[tid] = e0 * inv;
    lg[tid + 256] = e1 * inv;
    __syncthreads();

    // o_k[d] = sum_l prob[l] * embed_C(b,l,d); thread owns d = tid (coalesced)
    float ok = 0.0f;
    for (int l = 0; l < Llen; ++l) {
      float e = CC[(size_t)sidx[l * 4 + 0] * Dh + tid] +
                CC[(size_t)sidx[l * 4 + 1] * Dh + tid] +
                CC[(size_t)sidx[l * 4 + 2] * Dh + tid] +
                CC[(size_t)sidx[l * 4 + 3] * Dh + tid];
      const int rel = srel[l];
      if (rel >= 0) e += rnn_out[((size_t)b * Tlen + rel) * Dh + tid];
      ok = fmaf(lg[l], e, ok);
    }
    __syncthreads();
    u[tid] += ok;
    __syncthreads();
  }
  u_out[(size_t)b * Dh + tid] = u[tid];
}

// ---------------------------------------------------------------------------
// host-side orchestration
// ---------------------------------------------------------------------------
extern "C" void kernel_launch(void* const* d_in, const int* in_sizes, int n_in,
                              void* d_out, int out_size, void* d_ws,
                              size_t ws_size, hipStream_t stream) {
  const int* conv_seqs = (const int*)d_in[0];
  const int* src_seqs = (const int*)d_in[1];
  const int* kb_len = (const int*)d_in[2];
  const int* conv_len = (const int*)d_in[3];
  const float* emb_ctx = (const float*)d_in[4];
  const float* C_tables = (const float*)d_in[5];
  const float* Wih_f = (const float*)d_in[6];
  const float* Whh_f = (const float*)d_in[7];
  const float* bih_f = (const float*)d_in[8];
  const float* bhh_f = (const float*)d_in[9];
  const float* Wih_b = (const float*)d_in[10];
  const float* Whh_b = (const float*)d_in[11];
  const float* bih_b = (const float*)d_in[12];
  const float* bhh_b = (const float*)d_in[13];
  const float* W_w = (const float*)d_in[14];
  const float* W_b = (const float*)d_in[15];
  const float* proj_w = (const float*)d_in[16];
  const float* proj_b = (const float*)d_in[17];

  float* ws = (float*)d_ws;
  float* emb = ws;                                  // T*B*D   = 1048576
  float* gxf = emb + (size_t)Tlen * Bsz * Dh;       // T*B*3D  = 3145728
  float* gxb = gxf + (size_t)Tlen * Bsz * D3;
  float* ysf = gxb + (size_t)Tlen * Bsz * D3;       // T*B*D
  float* ysb = ysf + (size_t)Tlen * Bsz * Dh;       // T*B*D
  float* rnn_out = ysb + (size_t)Tlen * Bsz * Dh;   // B*T*D
  float* hid = rnn_out + (size_t)Bsz * Tlen * Dh;   // B*D
  float* uu = hid + (size_t)Bsz * Dh;               // B*D

  float* gp_out = (float*)d_out;                    // [B,L] = 16384
  float* enc_out = (float*)d_out + Bsz * Llen;      // [B,D] =  8192

  // 1) embedding bag for conversation
  embed_conv_kernel<<<Bsz * Tlen, Dh, 0, stream>>>(conv_seqs, emb_ctx, emb);

  // 2) input-gate GEMMs, both directions (12288 tiles, 8 waves/block)
  gx_gemm_kernel<<<dim3(1536, 2), 256, 0, stream>>>(emb, Wih_f, bih_f, Wih_b,
                                                    bih_b, gxf, gxb);

  // 3) sequential GRU, one block per direction, 224 KB dynamic LDS
  //    (h 32KB + gh 96KB + TDM-staged gx row 96KB; CDNA5 WGP has 320KB)
  gru_kernel<<<2, 512, (Bsz * Dh + 2 * Bsz * D3) * sizeof(float), stream>>>(
      gxf, gxb, Whh_f, bhh_f, Whh_b, bhh_b, ysf, ysb);

  // 4) rnn_out projection (4096 tiles)
  rnnout_gemm_kernel<<<512, 256, 0, stream>>>(ysf, ysb, W_w, W_b, rnn_out);

  // 5) hidden = cat(hT_f, hT_b) @ W_w^T + W_b
  //    hT_f = ys_f rows at t=T-1; hT_b = ys_b rows at t=0 (time-reversed store)
  cat_gemm32_kernel<<<4, 256, 0, stream>>>(
      ysf + (size_t)(Tlen - 1) * Bsz * Dh, ysb, W_w, W_b, hid, 0);

  // 6) fused 3-hop memory network -> global_pointer + final u
  hops_kernel<<<Bsz, 256, 0, stream>>>(src_seqs, kb_len, conv_len, C_tables,
                                       rnn_out, hid, uu, gp_out);

  // 7) encoded = relu(cat(hidden, u) @ proj_w^T + proj_b)
  cat_gemm32_kernel<<<4, 256, 0, stream>>>(hid, uu, proj_w, proj_b, enc_out, 1);
}